// ProtoNet_55499567399067
// MI455X (gfx1250) — compile-verified
//
#include <hip/hip_runtime.h>
#include <cstdint>
#include <cstddef>

// ---------------- problem constants ----------------
#define BS   256
#define NS   25
#define NQ   75
#define DIM  1024
#define NCLS 5
#define NN   100          // NS + NQ
#define NPAD 112          // 7 * 16 (tile-padded)
#define KCHK 64           // K chunk staged in LDS
#define SPITCH 68         // stage LDS pitch (mult of 4 -> 16B stores; 68%64=4 -> conflict-free frag reads)
#define LPITCH 116        // logits LDS pitch
#define TMAXV 4.605170185988092f

typedef __attribute__((ext_vector_type(2))) float v2f;
typedef __attribute__((ext_vector_type(8))) float v8f;
typedef __attribute__((ext_vector_type(4))) int   v4i;

typedef __attribute__((address_space(1))) v4i g_v4i;   // global
typedef __attribute__((address_space(3))) v4i l_v4i;   // LDS

// ---- CDNA5 async-copy helpers (ASYNCcnt-tracked LDS DMA, no VGPR bounce) ----
__device__ __forceinline__ void async_ld128(const float* g, float* l) {
#if __has_builtin(__builtin_amdgcn_global_load_async_to_lds_b128)
    __builtin_amdgcn_global_load_async_to_lds_b128(
        (g_v4i*)(v4i*)(void*)g,
        (l_v4i*)(v4i*)(void*)l, 0, 0);
#else
    asm volatile("global_load_async_to_lds_b128 %0, %1, off"
                 :: "v"((unsigned)(size_t)l), "v"(g) : "memory");
#endif
}
__device__ __forceinline__ void wait_async0() {
#if __has_builtin(__builtin_amdgcn_s_wait_asynccnt)
    __builtin_amdgcn_s_wait_asynccnt(0);
#else
    asm volatile("s_wait_asynccnt 0" ::: "memory");
#endif
}

// ---------------- kernel 1: inverse row norms of concat(img) / concat(txt) ----------------
__global__ __launch_bounds__(256) void k_norm(
    const float* __restrict__ sImg, const float* __restrict__ sTxt,
    const float* __restrict__ qImg, const float* __restrict__ qTxt,
    float* __restrict__ invIm, float* __restrict__ invTx)
{
    const int b = blockIdx.x;
    const int tid = threadIdx.x, wave = tid >> 5, lane = tid & 31;
    for (int r = wave; r < 2 * NPAD; r += 8) {
        const int kind = r / NPAD;     // 0=img 1=txt
        const int n = r % NPAD;
        float inv = 0.f;
        if (n < NN) {
            const float* base;
            if (kind == 0) base = (n < NS) ? sImg + ((size_t)b * NS + n) * DIM
                                           : qImg + ((size_t)b * NQ + (n - NS)) * DIM;
            else           base = (n < NS) ? sTxt + ((size_t)b * NS + n) * DIM
                                           : qTxt + ((size_t)b * NQ + (n - NS)) * DIM;
            const float4* v4 = (const float4*)base;
            float ss = 0.f;
            for (int j = lane; j < DIM / 4; j += 32) {
                float4 v = v4[j];
                ss += v.x * v.x + v.y * v.y + v.z * v.z + v.w * v.w;
            }
            #pragma unroll
            for (int off = 16; off; off >>= 1) ss += __shfl_xor(ss, off, 32);
            float r0 = rsqrtf(ss);
            inv = r0 * (1.5f - 0.5f * ss * r0 * r0);   // Newton refine
        }
        if (lane == 0) {
            if (kind == 0) invIm[b * NPAD + n] = inv;
            else           invTx[b * NPAD + n] = inv;
        }
    }
}

// ---------------- kernel 2: prototypes (scatter-mean over classes) ----------------
__global__ __launch_bounds__(256) void k_proto(
    const float* __restrict__ sImg, const float* __restrict__ sTxt,
    const int* __restrict__ sLab, float* __restrict__ protos)
{
    const int b = blockIdx.x;
    const int tid = threadIdx.x;                 // each thread owns 4 dims (float4)
    const float4* si = (const float4*)(sImg + (size_t)b * NS * DIM);
    const float4* st = (const float4*)(sTxt + (size_t)b * NS * DIM);
    float4 acc[NCLS] = {};
    int cnt[NCLS] = {};
    for (int s = 0; s < NS; ++s) {
        const int lbl = sLab[b * NS + s];
        float4 a = si[s * (DIM / 4) + tid];
        float4 c = st[s * (DIM / 4) + tid];
        const float fx = 0.5f * (a.x + c.x), fy = 0.5f * (a.y + c.y);
        const float fz = 0.5f * (a.z + c.z), fw = 0.5f * (a.w + c.w);
        #pragma unroll
        for (int cl = 0; cl < NCLS; ++cl) {
            if (lbl == cl) {
                acc[cl].x += fx; acc[cl].y += fy; acc[cl].z += fz; acc[cl].w += fw;
                cnt[cl]++;
            }
        }
    }
    #pragma unroll
    for (int cl = 0; cl < NCLS; ++cl) {
        const float inv = 1.f / fmaxf((float)cnt[cl], 1.f);
        float4 o;
        o.x = acc[cl].x * inv; o.y = acc[cl].y * inv;
        o.z = acc[cl].z * inv; o.w = acc[cl].w * inv;
        ((float4*)(protos + ((size_t)b * NCLS + cl) * DIM))[tid] = o;
    }
}

// ---------------- kernel 3: prototypical CE + accuracy ----------------
__global__ __launch_bounds__(256) void k_cls(
    const float* __restrict__ qImg, const float* __restrict__ qTxt,
    const float* __restrict__ protos, const int* __restrict__ qLab,
    float* __restrict__ clsP, float* __restrict__ accP)
{
    __shared__ float protoS[NCLS * DIM];
    __shared__ float redC[8], redA[8];
    const int b = blockIdx.x;
    const int tid = threadIdx.x, wave = tid >> 5, lane = tid & 31;

    const float4* psrc = (const float4*)(protos + (size_t)b * NCLS * DIM);
    for (int j = tid; j < NCLS * DIM / 4; j += 256) ((float4*)protoS)[j] = psrc[j];
    __syncthreads();

    float clsLocal = 0.f, accLocal = 0.f;
    for (int q = wave; q < NQ; q += 8) {
        const float4* qi = (const float4*)(qImg + ((size_t)b * NQ + q) * DIM);
        const float4* qt = (const float4*)(qTxt + ((size_t)b * NQ + q) * DIM);
        float dist[NCLS] = {0.f, 0.f, 0.f, 0.f, 0.f};
        for (int j = lane; j < DIM / 4; j += 32) {
            float4 a = qi[j], c = qt[j];
            const float fx = 0.5f * (a.x + c.x), fy = 0.5f * (a.y + c.y);
            const float fz = 0.5f * (a.z + c.z), fw = 0.5f * (a.w + c.w);
            #pragma unroll
            for (int cl = 0; cl < NCLS; ++cl) {
                const float4 p = ((const float4*)protoS)[cl * (DIM / 4) + j];
                const float dx = fx - p.x, dy = fy - p.y, dz = fz - p.z, dw = fw - p.w;
                dist[cl] += dx * dx + dy * dy + dz * dz + dw * dw;
            }
        }
        #pragma unroll
        for (int cl = 0; cl < NCLS; ++cl) {
            #pragma unroll
            for (int off = 16; off; off >>= 1) dist[cl] += __shfl_xor(dist[cl], off, 32);
        }
        if (lane == 0) {
            const int lbl = qLab[b * NQ + q];
            float mn = dist[0]; int am = 0;
            #pragma unroll
            for (int cl = 1; cl < NCLS; ++cl)
                if (dist[cl] < mn) { mn = dist[cl]; am = cl; }   // strict < : first-min tie (argmin)
            accLocal += (am == lbl) ? 1.f : 0.f;
            float se = 0.f;
            #pragma unroll
            for (int cl = 0; cl < NCLS; ++cl) se += expf(mn - dist[cl]);
            float dl = 0.f;
            #pragma unroll
            for (int cl = 0; cl < NCLS; ++cl) dl += (cl == lbl) ? dist[cl] : 0.f;
            // -log_softmax(-d)[lbl] = (d_lbl - mn) + log(sum exp(mn - d))
            clsLocal += (dl - mn) + logf(se);
        }
    }
    #pragma unroll
    for (int off = 16; off; off >>= 1) {
        clsLocal += __shfl_xor(clsLocal, off, 32);
        accLocal += __shfl_xor(accLocal, off, 32);
    }
    if (lane == 0) { redC[wave] = clsLocal; redA[wave] = accLocal; }
    __syncthreads();
    if (tid == 0) {
        float c = 0.f, a = 0.f;
        for (int w = 0; w < 8; ++w) { c += redC[w]; a += redA[w]; }
        clsP[b] = c; accP[b] = a;
    }
}

// ---------------- kernel 4: CLIP matching loss ----------------
// Double-buffered GLOBAL_LOAD_ASYNC_TO_LDS_B128 staging overlapped with the
// V_WMMA_F32_16X16X4_F32 Gram accumulation (112 WMMAs per chunk hide the DMA).
__device__ __forceinline__ void issue_chunk(
    const float* __restrict__ sImg, const float* __restrict__ sTxt,
    const float* __restrict__ qImg, const float* __restrict__ qTxt,
    int b, int kc, float* imgS, float* txtS, int tid)
{
    for (int j = tid; j < 2 * NN * (KCHK / 4); j += 256) {  // real rows only (0..99)
        const int mtx = (j >= NN * (KCHK / 4)) ? 1 : 0;
        const int id = mtx ? j - NN * (KCHK / 4) : j;
        const int row = id >> 4;            // KCHK/4 == 16 float4 per row
        const int c4 = id & 15;
        const float* base;
        if (mtx == 0) base = (row < NS) ? sImg + ((size_t)b * NS + row) * DIM
                                        : qImg + ((size_t)b * NQ + (row - NS)) * DIM;
        else          base = (row < NS) ? sTxt + ((size_t)b * NS + row) * DIM
                                        : qTxt + ((size_t)b * NQ + (row - NS)) * DIM;
        async_ld128(base + kc + c4 * 4, (mtx ? txtS : imgS) + row * SPITCH + c4 * 4);
    }
}

__global__ __launch_bounds__(256) void k_clip(
    const float* __restrict__ sImg, const float* __restrict__ sTxt,
    const float* __restrict__ qImg, const float* __restrict__ qTxt,
    const float* __restrict__ invIm, const float* __restrict__ invTx,
    const float* __restrict__ tPtr, float* __restrict__ clipP)
{
    // 2 ping-pong buffers x (img|txt) stage; logits buffer aliases buffer 0 afterwards
    __shared__ float smem[2 * 2 * NPAD * SPITCH];     // 121,856 B of the WGP's 320 KB
    __shared__ float invImS[NPAD], invTxS[NPAD];
    __shared__ float redS[8];
    float* logitsS = smem;                            // 112*116 floats < one buffer

    const int b = blockIdx.x;
    const int tid = threadIdx.x, wave = tid >> 5, lane = tid & 31;
    const int ln = lane & 15, hi = lane >> 4;

    if (tid < NPAD) {
        invImS[tid] = (tid < NN) ? invIm[b * NPAD + tid] : 0.f;
        invTxS[tid] = (tid < NN) ? invTx[b * NPAD + tid] : 0.f;
    }
    // zero the pad rows (100..111) of both matrices in both buffers, once
    for (int j = tid; j < 2 * 2 * 12 * (SPITCH / 4); j += 256) {
        int q = j;
        const int c4 = q % (SPITCH / 4); q /= (SPITCH / 4);
        const int row = NN + q % 12;     q /= 12;
        const int mtx = q % 2;           q /= 2;
        const int buf = q;
        float* dst = smem + buf * (2 * NPAD * SPITCH) + mtx * (NPAD * SPITCH)
                   + row * SPITCH + c4 * 4;
        *(float4*)dst = make_float4(0.f, 0.f, 0.f, 0.f);
    }

    v8f acc[7] = {};                     // up to 7 16x16 tiles per wave (49 tiles / 8 waves)

    // prologue: kick off chunk 0 into buffer 0
    issue_chunk(sImg, sTxt, qImg, qTxt, b, 0,
                smem, smem + NPAD * SPITCH, tid);

    int buf = 0;
    for (int kc = 0; kc < DIM; kc += KCHK, buf ^= 1) {
        wait_async0();                   // this wave's share of chunk kc is resident
        __syncthreads();                 // every wave's share resident; prev compute done
        if (kc + KCHK < DIM) {           // start DMA of the next chunk into the other buffer
            float* nb = smem + (buf ^ 1) * (2 * NPAD * SPITCH);
            issue_chunk(sImg, sTxt, qImg, qTxt, b, kc + KCHK,
                        nb, nb + NPAD * SPITCH, tid);
        }
        const float* imgS = smem + buf * (2 * NPAD * SPITCH);
        const float* txtS = imgS + NPAD * SPITCH;
        // 16 WMMA K-steps per tile; tile guard is wave-uniform -> EXEC stays all-ones
        #pragma unroll
        for (int i = 0; i < 7; ++i) {
            const int tile = wave + 8 * i;
            if (tile < 49) {
                const int tm = tile / 7, tn = tile % 7;
                const float* ar = imgS + (tm * 16 + ln) * SPITCH;
                const float* br = txtS + (tn * 16 + ln) * SPITCH;
                #pragma unroll
                for (int k = 0; k < KCHK; k += 4) {
                    // A 16x4 f32: VGPR0={K0 | K2}, VGPR1={K1 | K3} across lane halves; B mirrors with N per lane
                    v2f a, bb;
                    a.x = ar[k + 2 * hi]; a.y = ar[k + 2 * hi + 1];
                    bb.x = br[k + 2 * hi]; bb.y = br[k + 2 * hi + 1];
                    acc[i] = __builtin_amdgcn_wmma_f32_16x16x4_f32(
                        false, a, false, bb, (short)0, acc[i], false, false);
                }
            }
        }
    }
    __syncthreads();

    // epilogue: logits = exp(clip(t)) * inv|img| * inv|txt| * Gram -> LDS (C/D layout: M=r+8*hi, N=ln)
    const float scale = expf(fminf(fmaxf(tPtr[0], 0.f), TMAXV));
    #pragma unroll
    for (int i = 0; i < 7; ++i) {
        const int tile = wave + 8 * i;
        if (tile < 49) {
            const int tm = tile / 7, tn = tile % 7;
            const int col = tn * 16 + ln;
            const float it = invTxS[col] * scale;
            #pragma unroll
            for (int r = 0; r < 8; ++r) {
                const int row = tm * 16 + r + 8 * hi;
                logitsS[row * LPITCH + col] = acc[i][r] * it * invImS[row];
            }
        }
    }
    __syncthreads();

    // per-row and per-column LSE + diagonal
    float part = 0.f;
    if (tid < NN) {
        float mr = -INFINITY, mc = -INFINITY;
        for (int j = 0; j < NN; ++j) {
            mr = fmaxf(mr, logitsS[tid * LPITCH + j]);
            mc = fmaxf(mc, logitsS[j * LPITCH + tid]);
        }
        float sr = 0.f, sc = 0.f;
        for (int j = 0; j < NN; ++j) {
            sr += expf(logitsS[tid * LPITCH + j] - mr);
            sc += expf(logitsS[j * LPITCH + tid] - mc);
        }
        const float rlse = mr + logf(sr);
        const float clse = mc + logf(sc);
        part = 2.f * logitsS[tid * (LPITCH + 1)] - rlse - clse;
    }
    #pragma unroll
    for (int off = 16; off; off >>= 1) part += __shfl_xor(part, off, 32);
    if (lane == 0) redS[wave] = part;
    __syncthreads();
    if (tid == 0) {
        float s = 0.f;
        for (int w = 0; w < 8; ++w) s += redS[w];
        clipP[b] = s;
    }
}

// ---------------- kernel 5: deterministic final reduction ----------------
__global__ __launch_bounds__(256) void k_final(
    const float* __restrict__ accP, const float* __restrict__ clsP,
    const float* __restrict__ clipP, float* __restrict__ out)
{
    __shared__ float rA[8], rC[8], rM[8];
    const int tid = threadIdx.x, wave = tid >> 5, lane = tid & 31;
    float a = accP[tid], c = clsP[tid], m = clipP[tid];
    #pragma unroll
    for (int off = 16; off; off >>= 1) {
        a += __shfl_xor(a, off, 32);
        c += __shfl_xor(c, off, 32);
        m += __shfl_xor(m, off, 32);
    }
    if (lane == 0) { rA[wave] = a; rC[wave] = c; rM[wave] = m; }
    __syncthreads();
    if (tid == 0) {
        float A = 0.f, C = 0.f, M = 0.f;
        for (int w = 0; w < 8; ++w) { A += rA[w]; C += rC[w]; M += rM[w]; }
        out[0] = A / (float)(BS * NQ);                                  // accuracy
        out[1] = C / (float)(BS * NQ) + (-0.5f / (float)(BS * NN)) * M; // cls_loss + matching_loss
    }
}

// ---------------- host launcher ----------------
extern "C" void kernel_launch(void* const* d_in, const int* in_sizes, int n_in,
                              void* d_out, int out_size, void* d_ws, size_t ws_size,
                              hipStream_t stream) {
    (void)in_sizes; (void)n_in; (void)out_size; (void)ws_size;
    const float* sImg = (const float*)d_in[0];
    const float* sTxt = (const float*)d_in[1];
    const float* qImg = (const float*)d_in[2];
    const float* qTxt = (const float*)d_in[3];
    const int*   sLab = (const int*)d_in[4];
    const int*   qLab = (const int*)d_in[5];
    const float* tP   = (const float*)d_in[6];
    float* out = (float*)d_out;

    float* ws     = (float*)d_ws;
    float* invIm  = ws;                               // 256*112
    float* invTx  = invIm + (size_t)BS * NPAD;        // 256*112
    float* protos = invTx + (size_t)BS * NPAD;        // 256*5*1024
    float* clsP   = protos + (size_t)BS * NCLS * DIM; // 256
    float* accP   = clsP + BS;                        // 256
    float* clipP  = accP + BS;                        // 256

    k_norm <<<BS, 256, 0, stream>>>(sImg, sTxt, qImg, qTxt, invIm, invTx);
    k_proto<<<BS, 256, 0, stream>>>(sImg, sTxt, sLab, protos);
    k_cls  <<<BS, 256, 0, stream>>>(qImg, qTxt, protos, qLab, clsP, accP);
    k_clip <<<BS, 256, 0, stream>>>(sImg, sTxt, qImg, qTxt, invIm, invTx, tP, clipP);
    k_final<<<1, 256, 0, stream>>>(accP, clsP, clipP, out);
}